// SemanticDependencyLBP_5652176961693
// MI455X (gfx1250) — compile-verified
//
#include <hip/hip_runtime.h>
#include <hip/hip_bf16.h>
#include <stdint.h>

#define NEGV (-1e9f)
#define EPSV 1.1920929e-07f

constexpr int Bsz = 4;
constexpr int Nn  = 96;
constexpr int NN  = Nn * Nn;                     // 9216
constexpr long long N3   = (long long)NN * Nn;   // 884736
constexpr long long MSG  = 2LL * Bsz * N3;       // 7077888 elements per 5D tensor
constexpr long long LSTR = (long long)Bsz * N3;  // l-plane stride in 5D tensors
constexpr int QSZ = 2 * Bsz * NN;                // 73728
constexpr int QL  = Bsz * NN;                    // 36864 (l-plane stride in q)

typedef float v2f __attribute__((ext_vector_type(2)));
typedef float v8f __attribute__((ext_vector_type(8)));

struct alignas(16) f4 { float v[4]; };

__device__ __forceinline__ float lae(float x, float y) {
  float m = fmaxf(x, y);
  if (m < -1e8f) return m;
  float n = fminf(x, y);
  return m + log1pf(__expf(n - m));
}

__device__ __forceinline__ float wave_max(float v) {
#pragma unroll
  for (int o = 16; o > 0; o >>= 1) v = fmaxf(v, __shfl_xor(v, o, 32));
  return v;
}
__device__ __forceinline__ float wave_sum(float v) {
#pragma unroll
  for (int o = 16; o > 0; o >>= 1) v += __shfl_xor(v, o, 32);
  return v;
}

// ---------------- init: zero messages, q = e ----------------
__global__ void init_kernel(const float* __restrict__ s_edge,
                            float* m_sib, float* m_cop, float* m_grd,
                            float* qv, float* m_tree) {
  long long idx    = (long long)blockIdx.x * blockDim.x + threadIdx.x;
  long long stride = (long long)gridDim.x * blockDim.x;
  for (long long i = idx; i < MSG; i += stride) {
    m_sib[i] = 0.f; m_cop[i] = 0.f; m_grd[i] = 0.f;
  }
  for (long long i = idx; i < QSZ; i += stride) {
    m_tree[i] = 0.f;
    int l   = (int)(i / QL);
    int rem = (int)(i % QL);
    int b   = rem / NN;
    int hm  = rem % NN;
    int h = hm / Nn, m = hm % Nn;
    qv[i] = (l == 0) ? 0.f : s_edge[(long long)b * NN + m * Nn + h];  // e1[b,h,m]=s_edge[b,m,h]
  }
}

// ---------------- CRF scores: S[b,h,m] = (q1-mt1)-(q0-mt0) ----------------
__global__ void scores_kernel(const float* __restrict__ qv,
                              const float* __restrict__ m_tree,
                              float* __restrict__ Sc) {
  int idx = blockIdx.x * blockDim.x + threadIdx.x;
  if (idx >= QL) return;
  float v0 = qv[idx] - m_tree[idx];
  float v1 = qv[idx + QL] - m_tree[idx + QL];
  Sc[idx] = v1 - v0;
}

// ---------------- q transpose: qT[l,b,c,d] = q[l,b,d,c] ----------------
__global__ void transpose_q_kernel(const float* __restrict__ qv,
                                   float* __restrict__ qT) {
  int idx = blockIdx.x * blockDim.x + threadIdx.x;
  if (idx >= QSZ) return;
  int l   = idx / QL;
  int rem = idx % QL;
  int b   = rem / NN;
  int cd  = rem % NN;
  int c = cd / Nn, d = cd % Nn;
  qT[idx] = qv[(l * Bsz + b) * NN + d * Nn + c];
}

// ---------------- Eisner inside + outside (marginals) ----------------
// One workgroup per batch element; 16 wave32s; workgroup barriers per width step.
__global__ void __launch_bounds__(512) eisner_kernel(
    const float* __restrict__ Sc, const unsigned char* __restrict__ mask,
    float* __restrict__ tabs, float* __restrict__ bt) {
  const int b    = blockIdx.x;
  const int tid  = threadIdx.x;
  const int lane = tid & 31;
  const int wid  = tid >> 5;
  const int NW   = blockDim.x >> 5;

  float* CL  = tabs + (long long)b * 8 * NN;  // table[i][w]
  float* CR  = CL + NN;
  float* IL  = CR + NN;
  float* IR  = IL + NN;
  float* gCL = IR + NN;
  float* gCR = gCL + NN;
  float* gIL = gCR + NN;
  float* gIR = gIL + NN;
  const float* S = Sc + b * NN;   // S[h*Nn+m]
  float* Bm = bt + b * NN;        // marginal grad wrt s_hm

  __shared__ int s_len;
  if (tid == 0) {
    int len = 0;
    for (int m = 0; m < Nn; ++m) len += mask[(long long)b * NN + m * Nn + 0] ? 1 : 0;
    s_len = len;
  }
  for (int i = tid; i < NN; i += blockDim.x) {
    int w = i % Nn;
    float cinit = (w == 0) ? 0.f : NEGV;
    CL[i] = cinit; CR[i] = cinit;
    IL[i] = NEGV;  IR[i] = NEGV;
    gCL[i] = 0.f; gCR[i] = 0.f; gIL[i] = 0.f; gIR[i] = 0.f;
    Bm[i] = 0.f;
  }
  __threadfence_block();
  __syncthreads();
  const int len = s_len;
  if (tid == 0) gCR[len] = 1.0f;  // d logZ / d CR[0][len] = 1

  // ---- forward (inside) ----
  for (int w = 1; w < Nn; ++w) {
    for (int i = wid; i + w < Nn; i += NW) {
      int j = i + w;
      float mx = NEGV;
      for (int r = i + lane; r < j; r += 32)
        mx = fmaxf(mx, CR[i * Nn + (r - i)] + CL[(r + 1) * Nn + (j - r - 1)]);
      mx = wave_max(mx);
      float base;
      if (mx < -1e8f) base = NEGV;
      else {
        float s = 0.f;
        for (int r = i + lane; r < j; r += 32)
          s += __expf(CR[i * Nn + (r - i)] + CL[(r + 1) * Nn + (j - r - 1)] - mx);
        base = mx + __logf(wave_sum(s));
      }
      if (lane == 0) {
        IL[i * Nn + w] = base + S[j * Nn + i];   // I(j->i)
        IR[i * Nn + w] = base + S[i * Nn + j];   // I(i->j)
      }
    }
    __threadfence_block(); __syncthreads();
    for (int i = wid; i + w < Nn; i += NW) {
      int j = i + w;
      float mx = NEGV;
      for (int r = i + lane; r < j; r += 32)
        mx = fmaxf(mx, CL[i * Nn + (r - i)] + IL[r * Nn + (j - r)]);
      mx = wave_max(mx);
      float cl;
      if (mx < -1e8f) cl = NEGV;
      else {
        float s = 0.f;
        for (int r = i + lane; r < j; r += 32)
          s += __expf(CL[i * Nn + (r - i)] + IL[r * Nn + (j - r)] - mx);
        cl = mx + __logf(wave_sum(s));
      }
      float mx2 = NEGV;
      for (int r = i + 1 + lane; r <= j; r += 32)
        mx2 = fmaxf(mx2, IR[i * Nn + (r - i)] + CR[r * Nn + (j - r)]);
      mx2 = wave_max(mx2);
      float cr;
      if (mx2 < -1e8f) cr = NEGV;
      else {
        float s2 = 0.f;
        for (int r = i + 1 + lane; r <= j; r += 32)
          s2 += __expf(IR[i * Nn + (r - i)] + CR[r * Nn + (j - r)] - mx2);
        cr = mx2 + __logf(wave_sum(s2));
      }
      if (i == 0 && w != len) cr = NEGV;   // single-root constraint
      if (lane == 0) { CL[i * Nn + w] = cl; CR[i * Nn + w] = cr; }
    }
    __threadfence_block(); __syncthreads();
  }

  // ---- backward (outside / grad of logZ) ----
  for (int w = Nn - 1; w >= 1; --w) {
    // phase B1: distribute grads of CL/CR at width w
    for (int i = wid; i + w < Nn; i += NW) {
      int j = i + w;
      float crv = CR[i * Nn + w];
      float gcr = gCR[i * Nn + w];
      if (gcr != 0.f && crv > -1e8f) {
        for (int r = i + 1 + lane; r <= j; r += 32) {
          float wgt = __expf(IR[i * Nn + (r - i)] + CR[r * Nn + (j - r)] - crv) * gcr;
          gIR[i * Nn + (r - i)] += wgt;
          gCR[r * Nn + (j - r)] += wgt;
        }
      }
      float clv = CL[i * Nn + w];
      float gcl = gCL[i * Nn + w];
      if (gcl != 0.f && clv > -1e8f) {
        for (int r = i + lane; r < j; r += 32) {
          float wgt = __expf(CL[i * Nn + (r - i)] + IL[r * Nn + (j - r)] - clv) * gcl;
          gCL[i * Nn + (r - i)] += wgt;
          gIL[r * Nn + (j - r)] += wgt;
        }
      }
    }
    __threadfence_block(); __syncthreads();
    // phase B2: distribute grads of IL/IR at width w; emit arc marginals
    for (int i = wid; i + w < Nn; i += NW) {
      int j = i + w;
      float gil = gIL[i * Nn + w];
      float gir = gIR[i * Nn + w];
      if (lane == 0) {
        Bm[j * Nn + i] += gil;   // marginal of arc j->i (s_hm[j,i])
        Bm[i * Nn + j] += gir;   // marginal of arc i->j
      }
      float gb = gil + gir;
      float base = IL[i * Nn + w] - S[j * Nn + i];
      if (gb != 0.f && base > -1e8f) {
        for (int r = i + lane; r < j; r += 32) {
          float wgt = __expf(CR[i * Nn + (r - i)] + CL[(r + 1) * Nn + (j - r - 1)] - base) * gb;
          gCR[i * Nn + (r - i)] += wgt;
          gCL[(r + 1) * Nn + (j - r - 1)] += wgt;
        }
      }
    }
    __threadfence_block(); __syncthreads();
  }
}

// ---------------- tree message update (elementwise) ----------------
__global__ void tree_msg_kernel(const float* __restrict__ qv,
                                const float* __restrict__ bt,
                                float* __restrict__ m_tree) {
  int idx = blockIdx.x * blockDim.x + threadIdx.x;
  if (idx >= QL) return;
  float v0 = qv[idx] - m_tree[idx];
  float v1 = qv[idx + QL] - m_tree[idx + QL];
  float p  = bt[idx];
  float t0 = __logf(fmaxf(1.f - p, EPSV)) - v0;
  float t1 = __logf(fmaxf(p, EPSV)) - v1;
  float z  = lae(t0, t1);
  m_tree[idx]      = t0 - z;
  m_tree[idx + QL] = t1 - z;
}

// ---------------- fused sib/cop/grd message update + w2 weighting ----------------
// grid (24, Bsz); 96 threads = 4 'a' values x 24 d-chunks of 4; loop over c.
// All 5D traffic moves as b128. Writes new messages in place and
// WS[l,b,a,c,d] = (m_sib+m_cop+m_grd)*w2, which the WMMA kernel reduces over c.
__global__ void __launch_bounds__(96) msg_kernel(
    const float* __restrict__ s_sib, const float* __restrict__ s_cop,
    const float* __restrict__ s_grd, const unsigned char* __restrict__ mask,
    const float* __restrict__ qv, const float* __restrict__ qT,
    float* __restrict__ m_sib, float* __restrict__ m_cop,
    float* __restrict__ m_grd, float* __restrict__ WSb) {
  const int tid = threadIdx.x;
  const int dq  = tid % 24;             // d-chunk index
  const int as  = tid / 24;             // a sub-index (0..3)
  const int a   = blockIdx.x * 4 + as;
  const int b   = blockIdx.y;
  const int d0  = dq * 4;

  const f4 q0ad = *(const f4*)(qv + b * NN + a * Nn + d0);
  const f4 q1ad = *(const f4*)(qv + QL + b * NN + a * Nn + d0);
  float mda[4];
#pragma unroll
  for (int t = 0; t < 4; ++t) {
    int d = d0 + t;
    mda[t] = (mask[(long long)b * NN + d * Nn + a] != 0 && a != d) ? 1.f : 0.f;
  }

  for (int c = 0; c < Nn; ++c) {
    const long long mi = (((long long)b * Nn + a) * Nn + c) * Nn + d0;  // [b][a][c][d]
    const long long si = (((long long)b * Nn + c) * Nn + a) * Nn + d0;  // s_*[b][c][a][d]
    __builtin_prefetch(&s_sib[si + (long long)NN], 0, 0);
    __builtin_prefetch(&s_cop[si + (long long)NN], 0, 0);
    __builtin_prefetch(&s_grd[si + (long long)NN], 0, 0);

    const f4 ms0 = *(const f4*)(m_sib + mi);
    const f4 ms1 = *(const f4*)(m_sib + mi + LSTR);
    const f4 Ss  = *(const f4*)(s_sib + si);
    const f4 mc0 = *(const f4*)(m_cop + mi);
    const f4 mc1 = *(const f4*)(m_cop + mi + LSTR);
    const f4 Sc2 = *(const f4*)(s_cop + si);
    const f4 mg0 = *(const f4*)(m_grd + mi);
    const f4 mg1 = *(const f4*)(m_grd + mi + LSTR);
    const f4 Sg  = *(const f4*)(s_grd + si);
    const f4 qc0 = *(const f4*)(qT + b * NN + c * Nn + d0);        // q[0,b,d,c]
    const f4 qc1 = *(const f4*)(qT + QL + b * NN + c * Nn + d0);   // q[1,b,d,c]
    const f4 qg0 = *(const f4*)(qv + b * NN + c * Nn + d0);        // q[0,b,c,d]
    const f4 qg1 = *(const f4*)(qv + QL + b * NN + c * Nn + d0);   // q[1,b,c,d]
    const float mca =
        (mask[(long long)b * NN + c * Nn + a] != 0 && a != c) ? 1.f : 0.f;

    f4 ns0, ns1, nc0, nc1, ng0, ng1, ws0, ws1;
#pragma unroll
    for (int t = 0; t < 4; ++t) {
      // sibling: v[l] = q[l,b,a,d] - m_sib
      float v0 = q0ad.v[t] - ms0.v[t];
      float v1 = q1ad.v[t] - ms1.v[t];
      float t0 = lae(v0, v1), t1 = lae(v0, v1 + Ss.v[t]);
      float z  = lae(t0, t1);
      ns0.v[t] = t0 - z; ns1.v[t] = t1 - z;
      // co-parent: v[l] = q[l,b,d,c] - m_cop
      v0 = qc0.v[t] - mc0.v[t];
      v1 = qc1.v[t] - mc1.v[t];
      t0 = lae(v0, v1); t1 = lae(v0, v1 + Sc2.v[t]);
      z  = lae(t0, t1);
      nc0.v[t] = t0 - z; nc1.v[t] = t1 - z;
      // grandparent: v[l] = q[l,b,c,d] - m_grd
      v0 = qg0.v[t] - mg0.v[t];
      v1 = qg1.v[t] - mg1.v[t];
      t0 = lae(v0, v1); t1 = lae(v0, v1 + Sg.v[t]);
      z  = lae(t0, t1);
      ng0.v[t] = t0 - z; ng1.v[t] = t1 - z;

      float w2v = (c != d0 + t) ? mca * mda[t] : 0.f;
      ws0.v[t] = (ns0.v[t] + nc0.v[t] + ng0.v[t]) * w2v;
      ws1.v[t] = (ns1.v[t] + nc1.v[t] + ng1.v[t]) * w2v;
    }
    *(f4*)(m_sib + mi)        = ns0;
    *(f4*)(m_sib + mi + LSTR) = ns1;
    *(f4*)(m_cop + mi)        = nc0;
    *(f4*)(m_cop + mi + LSTR) = nc1;
    *(f4*)(m_grd + mi)        = ng0;
    *(f4*)(m_grd + mi + LSTR) = ng1;
    *(f4*)(WSb + mi)          = ws0;
    *(f4*)(WSb + mi + LSTR)   = ws1;
  }
}

// ---------------- WMMA reduction over c: agg[l,b,a,d] = sum_c WS[l,b,a,c,d] ----------------
// Stage a 96x16 [c][d] tile into LDS with coalesced b128 loads, then accumulate
// ones(16x4) x P(4x16) over 24 fully-unrolled K-chunks via V_WMMA_F32_16X16X4_F32,
// feeding B fragments from LDS. Every output row equals the column sum; row M=0 stored.
__global__ void __launch_bounds__(32) wmma_agg_kernel(const float* __restrict__ WSb,
                                                      float* __restrict__ agg) {
  __shared__ float tile[Nn * 16];     // 6 KB
  const int bx = blockIdx.x;
  const int dt = bx % 6;
  const int a  = (bx / 6) % Nn;
  const int b  = (bx / (6 * Nn)) % Bsz;
  const int l  = bx / (6 * Nn * Bsz);
  const int lane = threadIdx.x & 31;

  const float* P = WSb + ((long long)l * Bsz + b) * N3 + (long long)a * NN;  // [c][d]
  const int d0 = dt * 16;

  // cooperative load: 96 rows x 16 floats = 384 b128 chunks
#pragma unroll
  for (int t = 0; t < 12; ++t) {
    int ch = lane + 32 * t;           // 0..383
    int c  = ch >> 2;
    int f  = ch & 3;
    *(f4*)(tile + c * 16 + f * 4) = *(const f4*)(P + c * Nn + d0 + f * 4);
  }
  __syncthreads();

  const int kk = lane >> 4;   // K sub-row within B VGPR
  const int nc = lane & 15;   // N column
  v8f acc = {};
  v2f av; av.x = 1.0f; av.y = 1.0f;   // all-ones A matrix
#pragma unroll
  for (int kc = 0; kc < 24; ++kc) {
    const int cb = kc * 4;
    v2f bv;
    bv.x = tile[(cb + kk) * 16 + nc];
    bv.y = tile[(cb + 2 + kk) * 16 + nc];
    acc = __builtin_amdgcn_wmma_f32_16x16x4_f32(false, av, false, bv, (short)0, acc,
                                                false, false);
  }
  if (lane < 16)
    agg[((long long)l * Bsz + b) * NN + a * Nn + d0 + lane] = acc[0];
}

// ---------------- q = e + agg + m_tree ----------------
__global__ void q_update_kernel(const float* __restrict__ s_edge,
                                const float* __restrict__ agg,
                                const float* __restrict__ m_tree,
                                float* __restrict__ qv) {
  int idx = blockIdx.x * blockDim.x + threadIdx.x;
  if (idx >= QSZ) return;
  int l   = idx / QL;
  int rem = idx % QL;
  int b   = rem / NN;
  int hm  = rem % NN;
  int h = hm / Nn, m = hm % Nn;
  float e = (l == 0) ? 0.f : s_edge[b * NN + m * Nn + h];
  qv[idx] = e + agg[idx] + m_tree[idx];
}

// ---------------- output: softmax over l, take class 1 ----------------
__global__ void out_kernel(const float* __restrict__ qv, float* __restrict__ out) {
  int idx = blockIdx.x * blockDim.x + threadIdx.x;
  if (idx >= Bsz * NN) return;
  int b  = idx / NN;
  int mh = idx % NN;
  int m = mh / Nn, h = mh % Nn;
  float q0 = qv[b * NN + h * Nn + m];
  float q1 = qv[QL + b * NN + h * Nn + m];
  out[idx] = 1.f / (1.f + __expf(-(q1 - q0)));   // out[b,m,h]
}

extern "C" void kernel_launch(void* const* d_in, const int* in_sizes, int n_in,
                              void* d_out, int out_size, void* d_ws, size_t ws_size,
                              hipStream_t stream) {
  (void)in_sizes; (void)n_in; (void)out_size; (void)ws_size;
  const float* s_edge = (const float*)d_in[0];
  const float* s_sib  = (const float*)d_in[1];
  const float* s_cop  = (const float*)d_in[2];
  const float* s_grd  = (const float*)d_in[3];
  const unsigned char* mask = (const unsigned char*)d_in[4];
  float* out = (float*)d_out;

  float* w = (float*)d_ws;
  float* m_sib  = w; w += MSG;
  float* m_cop  = w; w += MSG;
  float* m_grd  = w; w += MSG;
  float* WSb    = w; w += MSG;
  float* qv     = w; w += QSZ;
  float* m_tree = w; w += QSZ;
  float* agg    = w; w += QSZ;
  float* qT     = w; w += QSZ;
  float* Sc     = w; w += Bsz * NN;
  float* bt     = w; w += Bsz * NN;
  float* tabs   = w; w += (long long)Bsz * 8 * NN;

  init_kernel<<<4096, 256, 0, stream>>>(s_edge, m_sib, m_cop, m_grd, qv, m_tree);
  for (int it = 0; it < 3; ++it) {
    scores_kernel<<<(QL + 255) / 256, 256, 0, stream>>>(qv, m_tree, Sc);
    eisner_kernel<<<Bsz, 512, 0, stream>>>(Sc, mask, tabs, bt);
    tree_msg_kernel<<<(QL + 255) / 256, 256, 0, stream>>>(qv, bt, m_tree);
    transpose_q_kernel<<<(QSZ + 255) / 256, 256, 0, stream>>>(qv, qT);
    msg_kernel<<<dim3(24, Bsz), 96, 0, stream>>>(s_sib, s_cop, s_grd, mask, qv, qT,
                                                 m_sib, m_cop, m_grd, WSb);
    wmma_agg_kernel<<<2 * Bsz * Nn * 6, 32, 0, stream>>>(WSb, agg);
    q_update_kernel<<<(QSZ + 255) / 256, 256, 0, stream>>>(s_edge, agg, m_tree, qv);
  }
  out_kernel<<<(Bsz * NN + 255) / 256, 256, 0, stream>>>(qv, out);
}